// actor_53781580480512
// MI455X (gfx1250) — compile-verified
//
#include <hip/hip_runtime.h>
#include <cstdint>
#include <cstddef>

#define ZDIM 64
#define PDIM 4096
#define TPB  256      // 8 wave32 per block
#define BT   32       // batch rows per LDS tile
#define NT   4        // b-tiles per block (double-buffered pipeline)
#define LN_EPS 1e-5f

#if __has_builtin(__builtin_amdgcn_tanhf)
__device__ __forceinline__ float fast_tanh(float x) { return __builtin_amdgcn_tanhf(x); }
#else
__device__ __forceinline__ float fast_tanh(float x) { return tanhf(x); }
#endif

#if __has_builtin(__builtin_amdgcn_rsqf)
__device__ __forceinline__ float fast_rsqrt(float x) { return __builtin_amdgcn_rsqf(x); }
#else
__device__ __forceinline__ float fast_rsqrt(float x) { return rsqrtf(x); }
#endif

// sigmoid(x) = 0.5 * tanh(0.5*x) + 0.5  -> one TRANS op on CDNA5
__device__ __forceinline__ float fast_sigmoid(float x) {
    return fmaf(0.5f, fast_tanh(0.5f * x), 0.5f);
}

typedef uint32_t v4u __attribute__((ext_vector_type(4)));
typedef int32_t  v8i __attribute__((ext_vector_type(8)));
typedef int32_t  v4i __attribute__((ext_vector_type(4)));

// Issue the async staging of one BT x 64 f32 tile (row stride 64) into LDS.
// TDM path: one wave issues a single tensor_load_to_lds (TENSORcnt).
// Fallback: per-lane global_load_async_to_lds_b128 (ASYNCcnt).
__device__ __forceinline__ void stage_tile(const float* gsrc, float* lds_dst,
                                           unsigned tid) {
#if __has_builtin(__builtin_amdgcn_tensor_load_to_lds)
    if (tid < 32) {
        uint64_t gaddr = (uint64_t)(uintptr_t)gsrc;
        uint32_t laddr = (uint32_t)(uintptr_t)lds_dst;     // LDS byte offset
        v4u g0;
        g0[0] = 1u;                                        // count=1 valid descriptor
        g0[1] = laddr;                                     // lds_addr  (bits 63:32)
        g0[2] = (uint32_t)(gaddr & 0xffffffffu);           // global_addr[31:0]
        g0[3] = (uint32_t)((gaddr >> 32) & 0x01ffffffu)    // global_addr[56:32]
              | (2u << 30);                                // type = 2 ("image")
        v8i g1v;
        g1v[0] = (int)(2u << 16);                          // wg_mask=0, data_size=4B
        g1v[1] = (int)((uint32_t)ZDIM << 16);              // tensor_dim0 = 64
        g1v[2] = (int)((uint32_t)BT << 16);                // tensor_dim1 = 32
        g1v[3] = (int)((uint32_t)ZDIM << 16);              // tile_dim0 = 64
        g1v[4] = (int)BT;                                  // tile_dim1 = 32, tile_dim2 = 0
        g1v[5] = (int)ZDIM;                                // tensor_dim0_stride = 64
        g1v[6] = 0;
        g1v[7] = 0;
        v4i z4 = {0, 0, 0, 0};
#if __clang_major__ >= 23
        v8i z8 = {0, 0, 0, 0, 0, 0, 0, 0};
        __builtin_amdgcn_tensor_load_to_lds(g0, g1v, z4, z4, z8, 0);
#else
        __builtin_amdgcn_tensor_load_to_lds(g0, g1v, z4, z4, 0);
#endif
    }
#else
    uint32_t lds0 = (uint32_t)(uintptr_t)lds_dst;
    uint32_t t16 = tid * 16u;
#pragma unroll
    for (int i = 0; i < (BT * ZDIM * 4) / (TPB * 16); ++i) {
        uint32_t goff = t16 + (uint32_t)i * (TPB * 16);
        uint32_t loff = lds0 + goff;
        asm volatile("global_load_async_to_lds_b128 %0, %1, %2"
                     :: "v"(loff), "v"(goff), "s"((const char*)gsrc) : "memory");
    }
#endif
}

__device__ __forceinline__ void wait_stage() {
#if __has_builtin(__builtin_amdgcn_tensor_load_to_lds)
#if __has_builtin(__builtin_amdgcn_s_wait_tensorcnt)
    __builtin_amdgcn_s_wait_tensorcnt((short)0);
#else
    asm volatile("s_wait_tensorcnt 0" ::: "memory");
#endif
#else
    asm volatile("s_wait_asynccnt 0" ::: "memory");
#endif
}

__global__ __launch_bounds__(TPB) void pairmlp_kernel(
    const float* __restrict__ x,    // [B, Z]
    const float* __restrict__ W1,   // [P, 2, 4]
    const float* __restrict__ b1,   // [P, 4]
    const float* __restrict__ g1,   // [P, 4]
    const float* __restrict__ be1,  // [P, 4]
    const float* __restrict__ W2,   // [P, 4, 4]
    const float* __restrict__ b2,   // [P, 4]
    const float* __restrict__ g2,   // [P, 4]
    const float* __restrict__ be2,  // [P, 4]
    const float* __restrict__ W3,   // [P, 4]
    const float* __restrict__ b3,   // [P]
    float* __restrict__ out)        // [B, P]
{
    __shared__ float xs[2][BT * ZDIM];   // 2 x 8 KB ping-pong x-tiles

    const int p     = blockIdx.x * TPB + threadIdx.x;  // this thread's pair id
    const int bbase = blockIdx.y * (BT * NT);          // first batch row of block
    const int z     = p >> 6;
    const int zp    = p & 63;

    // ---- Prologue: kick off DMA for tile 0, overlap with weight fetches.
    stage_tile(x + (size_t)bbase * ZDIM, &xs[0][0], threadIdx.x);

    // Per-thread weight registers: 53 floats, reused for all BT*NT batch rows
    // (whole 850 KB weight set is L2-resident on a 192 MB L2).
    float4 w1a = *(const float4*)(W1 + (size_t)p * 8);      // W1[p][0][:]
    float4 w1c = *(const float4*)(W1 + (size_t)p * 8 + 4);  // W1[p][1][:]
    float4 B1  = *(const float4*)(b1  + (size_t)p * 4);
    float4 G1  = *(const float4*)(g1  + (size_t)p * 4);
    float4 E1  = *(const float4*)(be1 + (size_t)p * 4);
    float4 w2r0 = *(const float4*)(W2 + (size_t)p * 16);
    float4 w2r1 = *(const float4*)(W2 + (size_t)p * 16 + 4);
    float4 w2r2 = *(const float4*)(W2 + (size_t)p * 16 + 8);
    float4 w2r3 = *(const float4*)(W2 + (size_t)p * 16 + 12);
    float4 B2  = *(const float4*)(b2  + (size_t)p * 4);
    float4 G2  = *(const float4*)(g2  + (size_t)p * 4);
    float4 E2  = *(const float4*)(be2 + (size_t)p * 4);
    float4 w3  = *(const float4*)(W3  + (size_t)p * 4);
    float  B3  = b3[p];

    wait_stage();
    __syncthreads();

#pragma unroll 1   // keep the tile loop rolled: compact I-cache footprint
    for (int t = 0; t < NT; ++t) {
        const float* xt = &xs[t & 1][0];

        // Prefetch tile t+1 into the other buffer; the end-of-iteration
        // barrier below guaranteed everyone finished reading it.
        if (t + 1 < NT)
            stage_tile(x + (size_t)(bbase + (t + 1) * BT) * ZDIM,
                       &xs[(t + 1) & 1][0], threadIdx.x);

        float* outp = out + (size_t)(bbase + t * BT) * PDIM + p;

#pragma unroll 2
        for (int bb = 0; bb < BT; ++bb) {
            const float a = xt[bb * ZDIM + z];   // wave-uniform broadcast read
            const float c = xt[bb * ZDIM + zp];  // lane-contiguous, conflict-free

            // ---- layer 1: h = tanh(pair @ W1 + b1)
            float h0 = fast_tanh(fmaf(a, w1a.x, fmaf(c, w1c.x, B1.x)));
            float h1 = fast_tanh(fmaf(a, w1a.y, fmaf(c, w1c.y, B1.y)));
            float h2 = fast_tanh(fmaf(a, w1a.z, fmaf(c, w1c.z, B1.z)));
            float h3 = fast_tanh(fmaf(a, w1a.w, fmaf(c, w1c.w, B1.w)));

            // ---- LayerNorm 1 over H=4 (biased variance)
            float m  = (h0 + h1 + h2 + h3) * 0.25f;
            float d0 = h0 - m, d1 = h1 - m, d2 = h2 - m, d3 = h3 - m;
            float v  = fmaf(d0, d0, fmaf(d1, d1, fmaf(d2, d2, d3 * d3))) * 0.25f;
            float r  = fast_rsqrt(v + LN_EPS);
            float n0 = fmaf(d0 * r, G1.x, E1.x);
            float n1 = fmaf(d1 * r, G1.y, E1.y);
            float n2 = fmaf(d2 * r, G1.z, E1.z);
            float n3 = fmaf(d3 * r, G1.w, E1.w);

            // ---- layer 2: u[k] = tanh(b2[k] + sum_h n[h] * W2[p][h][k])
            float u0 = fast_tanh(fmaf(n0, w2r0.x, fmaf(n1, w2r1.x, fmaf(n2, w2r2.x, fmaf(n3, w2r3.x, B2.x)))));
            float u1 = fast_tanh(fmaf(n0, w2r0.y, fmaf(n1, w2r1.y, fmaf(n2, w2r2.y, fmaf(n3, w2r3.y, B2.y)))));
            float u2 = fast_tanh(fmaf(n0, w2r0.z, fmaf(n1, w2r1.z, fmaf(n2, w2r2.z, fmaf(n3, w2r3.z, B2.z)))));
            float u3 = fast_tanh(fmaf(n0, w2r0.w, fmaf(n1, w2r1.w, fmaf(n2, w2r2.w, fmaf(n3, w2r3.w, B2.w)))));

            // ---- LayerNorm 2
            float m2 = (u0 + u1 + u2 + u3) * 0.25f;
            float e0 = u0 - m2, e1 = u1 - m2, e2 = u2 - m2, e3 = u3 - m2;
            float v2 = fmaf(e0, e0, fmaf(e1, e1, fmaf(e2, e2, e3 * e3))) * 0.25f;
            float r2 = fast_rsqrt(v2 + LN_EPS);
            float q0 = fmaf(e0 * r2, G2.x, E2.x);
            float q1 = fmaf(e1 * r2, G2.y, E2.y);
            float q2 = fmaf(e2 * r2, G2.z, E2.z);
            float q3 = fmaf(e3 * r2, G2.w, E2.w);

            // ---- layer 3 + sigmoid; stream write-once output past the caches
            float o = fmaf(q0, w3.x, fmaf(q1, w3.y, fmaf(q2, w3.z, fmaf(q3, w3.w, B3))));
            __builtin_nontemporal_store(fast_sigmoid(o), outp + (size_t)bb * PDIM);
        }

        // Close tile t+1's DMA and ensure all waves are done with buf[t&1]
        // before it gets overwritten by tile t+2's DMA next iteration.
        wait_stage();
        __syncthreads();
    }
}

extern "C" void kernel_launch(void* const* d_in, const int* in_sizes, int n_in,
                              void* d_out, int out_size, void* d_ws, size_t ws_size,
                              hipStream_t stream) {
    (void)n_in; (void)d_ws; (void)ws_size; (void)out_size;
    const float* x   = (const float*)d_in[0];
    const float* W1  = (const float*)d_in[1];
    const float* b1  = (const float*)d_in[2];
    const float* g1  = (const float*)d_in[3];
    const float* be1 = (const float*)d_in[4];
    const float* W2  = (const float*)d_in[5];
    const float* b2  = (const float*)d_in[6];
    const float* g2  = (const float*)d_in[7];
    const float* be2 = (const float*)d_in[8];
    const float* W3  = (const float*)d_in[9];
    const float* b3  = (const float*)d_in[10];
    float* out = (float*)d_out;

    const int B = in_sizes[0] / ZDIM;                       // 4096
    dim3 grid(PDIM / TPB, (B + BT * NT - 1) / (BT * NT));   // (16, 32) = 512 blocks
    dim3 block(TPB);
    pairmlp_kernel<<<grid, block, 0, stream>>>(x, W1, b1, g1, be1,
                                               W2, b2, g2, be2, W3, b3, out);
}